// Model_77515569758801
// MI455X (gfx1250) — compile-verified
//
#include <hip/hip_runtime.h>
#include <hip/hip_bf16.h>
#include <math.h>

typedef __attribute__((ext_vector_type(2))) float v2f;
typedef __attribute__((ext_vector_type(8))) float v8f;

#define NB1 512  // stage-1 BN stat blocks

// ---------------- embed: x[i] = emb[nodes[i]] (N x 32) ----------------
__global__ void k_embed(const int* __restrict__ nodes, const float* __restrict__ emb,
                        float* __restrict__ x, int n) {
  int i = blockIdx.x * blockDim.x + threadIdx.x;
  if (i >= n) return;
  int t = nodes[i];
  const float4* s = (const float4*)(emb + (size_t)t * 32);
  float4* d = (float4*)(x + (size_t)i * 32);
#pragma unroll
  for (int j = 0; j < 8; ++j) d[j] = s[j];
}

// ---------------- zero fill ----------------
__global__ void k_fill0(float4* __restrict__ p, int cnt4) {
  int i = blockIdx.x * blockDim.x + threadIdx.x;
  if (i < cnt4) p[i] = make_float4(0.f, 0.f, 0.f, 0.f);
}

// ---------------- BN stats stage 1: per-block partial sum/sumsq per channel ----------------
__global__ __launch_bounds__(256) void k_bnstats1(const float* __restrict__ X,
                                                  float* __restrict__ partials, int n) {
  __shared__ float sh[512];
  int tid = threadIdx.x;
  int chan = tid & 31;
  int rg = tid >> 5;  // 0..7
  float sum = 0.f, sq = 0.f;
  for (int r = blockIdx.x * 8 + rg; r < n; r += gridDim.x * 8) {
    float v = X[(size_t)r * 32 + chan];
    sum += v;
    sq += v * v;
  }
  sh[tid] = sum;
  sh[256 + tid] = sq;
  __syncthreads();
  if (tid < 32) {
    float s0 = 0.f, q0 = 0.f;
#pragma unroll
    for (int g = 0; g < 8; ++g) {
      s0 += sh[g * 32 + tid];
      q0 += sh[256 + g * 32 + tid];
    }
    partials[blockIdx.x * 64 + tid] = s0;
    partials[blockIdx.x * 64 + 32 + tid] = q0;
  }
}

// ---------------- BN stats stage 2: deterministic fixed-order reduce ----------------
__global__ void k_bnstats2(const float* __restrict__ partials, float* __restrict__ stats, int nb) {
  int tid = threadIdx.x;  // 64 threads
  float acc = 0.f;
  for (int b = 0; b < nb; ++b) acc += partials[b * 64 + tid];
  stats[tid] = acc;
}

// ---------------- fused (BN+ReLU) + GEMM  Y[N x 16] = act(X[N x 32]) @ W[32 x 16] ----------------
// One wave computes a 16x16 output tile with 8 chained V_WMMA_F32_16X16X4_F32 (K=32).
__global__ __launch_bounds__(256) void k_gemm32x16(
    const float* __restrict__ X, const float* __restrict__ W, float* __restrict__ Y,
    const float* __restrict__ stats, const float* __restrict__ gamma,
    const float* __restrict__ beta, int use_bn, float invn, int n) {
  int gtid = blockIdx.x * blockDim.x + threadIdx.x;
  int wave = gtid >> 5;
  int lane = threadIdx.x & 31;
  int r0 = wave * 16;
  if (r0 >= n) return;  // wave-uniform: EXEC stays all-ones for WMMA
  int hi = lane >> 4, ln = lane & 15;
  int row = r0 + ln;  // n is a multiple of 16

  v2f a[8], b[8];
#pragma unroll
  for (int k = 0; k < 8; ++k) {
    int c = 4 * k + 2 * hi;  // A layout: VGPR0 holds K=c, VGPR1 holds K=c+1 for this lane
    v2f xv = *(const v2f*)(X + (size_t)row * 32 + c);
    if (use_bn) {
      float m0 = stats[c] * invn;
      float m1 = stats[c + 1] * invn;
      float v0 = fmaxf(stats[32 + c] * invn - m0 * m0, 0.f);
      float v1 = fmaxf(stats[32 + c + 1] * invn - m1 * m1, 0.f);
      float s0 = gamma[c] * rsqrtf(v0 + 1e-5f);
      float s1 = gamma[c + 1] * rsqrtf(v1 + 1e-5f);
      xv.x = fmaxf((xv.x - m0) * s0 + beta[c], 0.f);
      xv.y = fmaxf((xv.y - m1) * s1 + beta[c + 1], 0.f);
    }
    a[k] = xv;
    // B layout (4x16): VGPR0 = row K=c (lanes 0-15 / c+2 hi-half), VGPR1 = row K=c+1
    b[k].x = W[c * 16 + ln];
    b[k].y = W[(c + 1) * 16 + ln];
  }
  v8f acc = {};
#pragma unroll
  for (int k = 0; k < 8; ++k)
    acc = __builtin_amdgcn_wmma_f32_16x16x4_f32(false, a[k], false, b[k], (short)0, acc,
                                                false, false);
#pragma unroll
  for (int v = 0; v < 8; ++v)
    Y[(size_t)(r0 + v + 8 * hi) * 16 + ln] = acc[v];  // D: VGPR v -> rows v (lanes 0-15), v+8
}

// ---------------- edge scatter: acc[dst][0:16] += yo[src]; acc[src][16:32] += yb[dst] ----------------
__global__ __launch_bounds__(256) void k_scatter(const float* __restrict__ yo,
                                                 const float* __restrict__ yb,
                                                 const int* __restrict__ src,
                                                 const int* __restrict__ dst,
                                                 float* __restrict__ acc, int e) {
  int t = blockIdx.x * blockDim.x + threadIdx.x;
  int ei = t >> 2;
  if (ei >= e) return;
  int g = (t & 3) * 4;
  int s = src[ei], d = dst[ei];
  float4 vo = *(const float4*)(yo + (size_t)s * 16 + g);
  float4 vb = *(const float4*)(yb + (size_t)d * 16 + g);
  float* po = acc + (size_t)d * 32 + g;       // "out" half, cols 0..15
  float* pb = acc + (size_t)s * 32 + 16 + g;  // "back" half, cols 16..31
  atomicAdd(po + 0, vo.x);
  atomicAdd(po + 1, vo.y);
  atomicAdd(po + 2, vo.z);
  atomicAdd(po + 3, vo.w);
  atomicAdd(pb + 0, vb.x);
  atomicAdd(pb + 1, vb.y);
  atomicAdd(pb + 2, vb.z);
  atomicAdd(pb + 3, vb.w);
}

// ---------------- head: s = relu(relu(x[idx]) @ wh + bh) @ wo + bo ----------------
// One wave: 16 selected rows; 64 column tiles of wh, 8 WMMAs each; fused bias/relu/wo dot.
__global__ __launch_bounds__(256) void k_head(const float* __restrict__ X,
                                              const int* __restrict__ indices,
                                              const float* __restrict__ wh,
                                              const float* __restrict__ bh,
                                              const float* __restrict__ wo,
                                              const float* __restrict__ bo,
                                              float* __restrict__ s, int m, int H) {
  int gtid = blockIdx.x * blockDim.x + threadIdx.x;
  int wave = gtid >> 5;
  int lane = threadIdx.x & 31;
  int r0 = wave * 16;
  if (r0 >= m) return;
  int hi = lane >> 4, ln = lane & 15;
  int ridx = r0 + ln;
  int row = indices[ridx < m ? ridx : (m - 1)];

  v2f a[8];
#pragma unroll
  for (int k = 0; k < 8; ++k) {
    int c = 4 * k + 2 * hi;
    v2f xv = *(const v2f*)(X + (size_t)row * 32 + c);
    xv.x = fmaxf(xv.x, 0.f);
    xv.y = fmaxf(xv.y, 0.f);
    a[k] = xv;
  }
  float racc[8];
#pragma unroll
  for (int v = 0; v < 8; ++v) racc[v] = 0.f;

  int ntiles = H >> 4;
  for (int t = 0; t < ntiles; ++t) {
    int col = t * 16 + ln;
    v8f acc = {};
#pragma unroll
    for (int k = 0; k < 8; ++k) {
      int kr = 4 * k + 2 * hi;
      v2f b;
      b.x = wh[(size_t)kr * H + col];
      b.y = wh[(size_t)(kr + 1) * H + col];
      acc = __builtin_amdgcn_wmma_f32_16x16x4_f32(false, a[k], false, b, (short)0, acc,
                                                  false, false);
    }
    float bcol = bh[col], wcol = wo[col];
#pragma unroll
    for (int v = 0; v < 8; ++v) {
      float hval = fmaxf(acc[v] + bcol, 0.f);
      racc[v] += hval * wcol;
    }
  }
  // reduce over the 16 lanes (columns) within each half-wave
#pragma unroll
  for (int v = 0; v < 8; ++v) {
    float r = racc[v];
    r += __shfl_xor(r, 1);
    r += __shfl_xor(r, 2);
    r += __shfl_xor(r, 4);
    r += __shfl_xor(r, 8);
    racc[v] = r;
  }
  if (ln == 0) {
    float bias = bo[0];
#pragma unroll
    for (int v = 0; v < 8; ++v) {
      int r = r0 + v + 8 * hi;
      if (r < m) s[r] = racc[v] + bias;
    }
  }
}

// ---------------- deterministic per-segment max + log-sum-exp (block per segment) ----------------
__global__ __launch_bounds__(256) void k_segred(const float* __restrict__ s,
                                                const int* __restrict__ assignment,
                                                const int* __restrict__ indices,
                                                float* __restrict__ mx,
                                                float* __restrict__ lse, int m) {
  int seg = blockIdx.x;
  int tid = threadIdx.x;
  __shared__ float red[256];
  float lmax = -INFINITY;
  for (int i = tid; i < m; i += 256)
    if (assignment[indices[i]] == seg) lmax = fmaxf(lmax, s[i]);
  red[tid] = lmax;
  __syncthreads();
  for (int o = 128; o > 0; o >>= 1) {
    if (tid < o) red[tid] = fmaxf(red[tid], red[tid + o]);
    __syncthreads();
  }
  float gmax = red[0];
  __syncthreads();
  float lsum = 0.f;
  for (int i = tid; i < m; i += 256)
    if (assignment[indices[i]] == seg) lsum += expf(s[i] - gmax);
  red[tid] = lsum;
  __syncthreads();
  for (int o = 128; o > 0; o >>= 1) {
    if (tid < o) red[tid] += red[tid + o];
    __syncthreads();
  }
  if (tid == 0) {
    mx[seg] = gmax;
    lse[seg] = logf(red[0]);
  }
}

__global__ void k_finish(const float* __restrict__ s, const int* __restrict__ assignment,
                         const int* __restrict__ indices, const float* __restrict__ mx,
                         const float* __restrict__ lse, float* __restrict__ out, int m) {
  int i = blockIdx.x * blockDim.x + threadIdx.x;
  if (i >= m) return;
  int seg = assignment[indices[i]];
  out[i] = s[i] - mx[seg] - lse[seg];
}

// ---------------- host ----------------
extern "C" void kernel_launch(void* const* d_in, const int* in_sizes, int n_in,
                              void* d_out, int out_size, void* d_ws, size_t ws_size,
                              hipStream_t stream) {
  const int* assignment = (const int*)d_in[1];
  const int* nodes = (const int*)d_in[2];
  const int* src = (const int*)d_in[3];
  const int* dst = (const int*)d_in[4];
  const int* indices = (const int*)d_in[5];
  const float* emb = (const float*)d_in[6];
  const float* w0_out = (const float*)d_in[7];
  const float* w0_back = (const float*)d_in[8];
  const float* bn1_g = (const float*)d_in[9];
  const float* bn1_b = (const float*)d_in[10];
  const float* w1_out = (const float*)d_in[11];
  const float* w1_back = (const float*)d_in[12];
  const float* bn2_g = (const float*)d_in[13];
  const float* bn2_b = (const float*)d_in[14];
  const float* w2_out = (const float*)d_in[15];
  const float* w2_back = (const float*)d_in[16];
  const float* wh = (const float*)d_in[17];
  const float* bh = (const float*)d_in[18];
  const float* wo = (const float*)d_in[19];
  const float* bo = (const float*)d_in[20];

  const int n = in_sizes[2];
  const int e = in_sizes[3];
  const int m = in_sizes[5];
  const int H = in_sizes[18];
  const int L = in_sizes[9] / 32;
  const float invn = 1.0f / (float)n;

  // workspace layout (floats)
  float* w = (float*)d_ws;
  float* bufA = w;
  float* bufB = bufA + (size_t)n * 32;
  float* bufC = bufB + (size_t)n * 32;
  float* yo = bufC + (size_t)n * 32;
  float* yb = yo + (size_t)n * 16;
  float* sv = yb + (size_t)n * 16;
  float* partials = sv + m;
  float* stats = partials + (size_t)64 * NB1;
  float* mxb = stats + 64;
  float* lseb = mxb + 32;

  const int B = 256;
  int g_embed = (n + B - 1) / B;
  int g_fill = ((n * 8) + B - 1) / B;  // n*32 floats as float4
  int g_gemm = ((n * 2) + B - 1) / B;  // (n/16) waves * 32 lanes
  int g_scat = ((e * 4) + B - 1) / B;
  int g_head = ((m * 2) + B - 1) / B;
  int g_fin = (m + B - 1) / B;

  // ---- x0 = biconv(emb[nodes], w0_out, w0_back) ----
  k_embed<<<g_embed, B, 0, stream>>>(nodes, emb, bufC, n);
  k_gemm32x16<<<g_gemm, B, 0, stream>>>(bufC, w0_out, yo, stats, stats, stats, 0, invn, n);
  k_gemm32x16<<<g_gemm, B, 0, stream>>>(bufC, w0_back, yb, stats, stats, stats, 0, invn, n);
  k_fill0<<<g_fill, B, 0, stream>>>((float4*)bufA, n * 8);
  k_scatter<<<g_scat, B, 0, stream>>>(yo, yb, src, dst, bufA, e);

  float* x = bufA;
  float* alt = bufC;
  for (int l = 0; l < L; ++l) {
    // h = biconv(relu(bn1(x)), w1_out[l], w1_back[l])
    k_bnstats1<<<NB1, B, 0, stream>>>(x, partials, n);
    k_bnstats2<<<1, 64, 0, stream>>>(partials, stats, NB1);
    k_gemm32x16<<<g_gemm, B, 0, stream>>>(x, w1_out + (size_t)l * 512, yo, stats,
                                          bn1_g + l * 32, bn1_b + l * 32, 1, invn, n);
    k_gemm32x16<<<g_gemm, B, 0, stream>>>(x, w1_back + (size_t)l * 512, yb, stats,
                                          bn1_g + l * 32, bn1_b + l * 32, 1, invn, n);
    k_fill0<<<g_fill, B, 0, stream>>>((float4*)bufB, n * 8);
    k_scatter<<<g_scat, B, 0, stream>>>(yo, yb, src, dst, bufB, e);
    // x = x + biconv(relu(bn2(h)), w2_out[l], w2_back[l])  (scatter into a copy of x)
    k_bnstats1<<<NB1, B, 0, stream>>>(bufB, partials, n);
    k_bnstats2<<<1, 64, 0, stream>>>(partials, stats, NB1);
    k_gemm32x16<<<g_gemm, B, 0, stream>>>(bufB, w2_out + (size_t)l * 512, yo, stats,
                                          bn2_g + l * 32, bn2_b + l * 32, 1, invn, n);
    k_gemm32x16<<<g_gemm, B, 0, stream>>>(bufB, w2_back + (size_t)l * 512, yb, stats,
                                          bn2_g + l * 32, bn2_b + l * 32, 1, invn, n);
    hipMemcpyAsync(alt, x, (size_t)n * 32 * sizeof(float), hipMemcpyDeviceToDevice, stream);
    k_scatter<<<g_scat, B, 0, stream>>>(yo, yb, src, dst, alt, e);
    float* tmp = x;
    x = alt;
    alt = tmp;
  }

  // ---- head + segmented log-softmax ----
  k_head<<<g_head, B, 0, stream>>>(x, indices, wh, bh, wo, bo, sv, m, H);
  k_segred<<<32, B, 0, stream>>>(sv, assignment, indices, mxb, lseb, m);
  k_finish<<<g_fin, B, 0, stream>>>(sv, assignment, indices, mxb, lseb, (float*)d_out, m);
}